// PyGModel_81999515615309
// MI455X (gfx1250) — compile-verified
//
#include <hip/hip_runtime.h>
#include <math.h>

#define N_NODES    200000
#define N_EDGES    1250000
#define HID        64
#define NUM_GRAPHS 128

typedef float v2f __attribute__((ext_vector_type(2)));
typedef float v8f __attribute__((ext_vector_type(8)));

// ---- monotonic float<->uint encoding for atomic max over signed floats ----
__device__ __forceinline__ unsigned f2ord(float f) {
  unsigned u = __float_as_uint(f);
  return (u & 0x80000000u) ? ~u : (u | 0x80000000u);
}
__device__ __forceinline__ float ord2f(unsigned e) {
  return __uint_as_float((e & 0x80000000u) ? (e ^ 0x80000000u) : ~e);
}

// ---------------- edge scatter-add: agg[dst] += h[src] ----------------
// 32 threads per edge, float2 per lane: edge index is wave-uniform, loads are b64.
// Node table (51.2 MB) is L2-resident (192 MB L2), so gathers/atomics stay on-chip.
__global__ void __launch_bounds__(256) scatter_add_kernel(
    const float* __restrict__ h, const int* __restrict__ src,
    const int* __restrict__ dst, float* __restrict__ agg) {
  unsigned t = blockIdx.x * 256u + threadIdx.x;
  unsigned e = t >> 5, dp = (t & 31u) * 2u;
  if (e < N_EDGES) {
    int s  = src[e];
    int dd = dst[e];
    const float2 hv = *(const float2*)&h[(size_t)s * HID + dp];
    atomicAdd(&agg[(size_t)dd * HID + dp + 0], hv.x);
    atomicAdd(&agg[(size_t)dd * HID + dp + 1], hv.y);
  }
}

// ---------------- fused GIN MLP: out = [relu]( relu((h+agg)@W1+b1) @ W2 + b2 ) ----
// Block = 128 threads = 4 waves; block covers 64 nodes (16 per wave).
// Uses V_WMMA_F32_16X16X4_F32: A=16x4 f32 (v2f), B=4x16 f32 (v2f), C/D=16x16 f32 (v8f).
// A layout: lane m (0-15): VGPR0/1 = K={0,1}; lane m+16: K={2,3}.
// B layout: lane m: rows K={0,1}; lane m+16: rows K={2,3}; N = lane%16.
// C/D layout: vgpr v, lane<16 -> (M=v, N=lane); lane>=16 -> (M=v+8, N=lane-16).
__global__ void __launch_bounds__(128) gin_mlp_kernel(
    const float* __restrict__ hin, const float* __restrict__ agg,
    const float* __restrict__ W1, const float* __restrict__ B1,
    const float* __restrict__ W2, const float* __restrict__ B2,
    float* __restrict__ out, int relu_out) {
  // pitch 68 floats = 272 B: conflict-free column access AND 16B-aligned rows
  __shared__ float zbuf[64][68];
  __shared__ float tbuf[64][68];

  const int tid = threadIdx.x;
  const unsigned nodeBase = blockIdx.x * 64u;

  // Phase A: z = h + agg staged into LDS via b128 loads (64 nodes x 16 float4)
  for (int i = 0; i < 8; ++i) {
    int idx = tid + i * 128;           // 0..1023
    int nl = idx >> 4, c4 = (idx & 15) * 4;
    size_t g = (size_t)(nodeBase + (unsigned)nl) * HID + (unsigned)c4;
    float4 a = *(const float4*)&hin[g];
    float4 b = *(const float4*)&agg[g];
    float4 z;
    z.x = a.x + b.x; z.y = a.y + b.y; z.z = a.z + b.z; z.w = a.w + b.w;
    *(float4*)&zbuf[nl][c4] = z;
  }
  __syncthreads();

  const int wave = tid >> 5, lane = tid & 31;
  const int m = lane & 15, hi = lane >> 4;
  const int rowBase = wave * 16;

  v8f acc[4];

  // ---- GEMM1: t = relu(z @ W1 + b1) ----
  for (int j = 0; j < 4; ++j) {
    float b = B1[j * 16 + m];
    for (int v = 0; v < 8; ++v) acc[j][v] = b;
  }
  for (int k0 = 0; k0 < HID; k0 += 4) {
    v2f a = *(const v2f*)&zbuf[rowBase + m][k0 + hi * 2];
    for (int j = 0; j < 4; ++j) {
      int n = j * 16 + m;
      int k = k0 + hi * 2;
      v2f b;
      b.x = W1[(size_t)k * HID + n];
      b.y = W1[(size_t)(k + 1) * HID + n];
      acc[j] = __builtin_amdgcn_wmma_f32_16x16x4_f32(
          false, a, false, b, (short)0, acc[j], false, false);
    }
  }
  for (int j = 0; j < 4; ++j)
    for (int v = 0; v < 8; ++v) {
      float t = acc[j][v];
      tbuf[rowBase + v + hi * 8][j * 16 + m] = t > 0.f ? t : 0.f;
    }
  __syncthreads();

  // ---- GEMM2: out = [relu](t @ W2 + b2) ----
  for (int j = 0; j < 4; ++j) {
    float b = B2[j * 16 + m];
    for (int v = 0; v < 8; ++v) acc[j][v] = b;
  }
  for (int k0 = 0; k0 < HID; k0 += 4) {
    v2f a = *(const v2f*)&tbuf[rowBase + m][k0 + hi * 2];
    for (int j = 0; j < 4; ++j) {
      int n = j * 16 + m;
      int k = k0 + hi * 2;
      v2f b;
      b.x = W2[(size_t)k * HID + n];
      b.y = W2[(size_t)(k + 1) * HID + n];
      acc[j] = __builtin_amdgcn_wmma_f32_16x16x4_f32(
          false, a, false, b, (short)0, acc[j], false, false);
    }
  }
  // stage result tile back into zbuf (free after GEMM1), then coalesced b128 stores
  for (int j = 0; j < 4; ++j)
    for (int v = 0; v < 8; ++v) {
      float t = acc[j][v];
      if (relu_out) t = t > 0.f ? t : 0.f;
      zbuf[rowBase + v + hi * 8][j * 16 + m] = t;
    }
  __syncthreads();
  for (int i = 0; i < 8; ++i) {
    int idx = tid + i * 128;
    int nl = idx >> 4, c4 = (idx & 15) * 4;
    float4 r = *(const float4*)&zbuf[nl][c4];
    *(float4*)&out[(size_t)(nodeBase + (unsigned)nl) * HID + (unsigned)c4] = r;
  }
}

// ---------------- pooling ----------------
__global__ void __launch_bounds__(256) pool_init_kernel(float* psum, unsigned* pmax) {
  int i = blockIdx.x * 256 + threadIdx.x;
  if (i < NUM_GRAPHS * HID) {
    psum[i] = 0.f;
    pmax[i] = 0x007FFFFFu;  // f2ord(-inf)
  }
}

__global__ void __launch_bounds__(256) pool_kernel(
    const float* __restrict__ h, const int* __restrict__ batch,
    float* __restrict__ psum, unsigned* __restrict__ pmax) {
  unsigned idx = blockIdx.x * 256u + threadIdx.x;
  if (idx < (unsigned)N_NODES * HID) {
    unsigned node = idx >> 6, d = idx & 63u;
    float v = h[idx];
    int g = batch[node];
    atomicAdd(&psum[(size_t)g * HID + d], v);
    atomicMax(&pmax[(size_t)g * HID + d], f2ord(v));
  }
}

__global__ void __launch_bounds__(256) pool_combine_kernel(
    const float* __restrict__ psum, const unsigned* __restrict__ pmax,
    float* __restrict__ pooled) {
  int i = blockIdx.x * 256 + threadIdx.x;  // NUM_GRAPHS * 2*HID
  if (i < NUM_GRAPHS * 2 * HID) {
    int g = i >> 7, c = i & 127;
    pooled[i] = (c < HID) ? psum[g * HID + c] : ord2f(pmax[g * HID + (c - HID)]);
  }
}

// ---------------- tiny dense head layer (SiLU or linear) ----------------
__global__ void dense_kernel(const float* __restrict__ in, const float* __restrict__ W,
                             const float* __restrict__ B, float* __restrict__ out,
                             int din, int dout, int act) {
  int g = blockIdx.x, j = threadIdx.x;
  if (j < dout) {
    float acc = B[j];
    for (int i = 0; i < din; ++i) acc += in[g * din + i] * W[i * dout + j];
    if (act) acc = acc / (1.f + __expf(-acc));  // SiLU
    out[g * dout + j] = acc;
  }
}

extern "C" void kernel_launch(void* const* d_in, const int* in_sizes, int n_in,
                              void* d_out, int out_size, void* d_ws, size_t ws_size,
                              hipStream_t stream) {
  (void)in_sizes; (void)n_in; (void)out_size; (void)ws_size;

  const float* x     = (const float*)d_in[0];
  const int*   ei    = (const int*)d_in[1];
  const int*   src   = ei;
  const int*   dst   = ei + N_EDGES;
  const int*   batch = (const int*)d_in[2];
  const float* c1w1  = (const float*)d_in[3];
  const float* c1b1  = (const float*)d_in[4];
  const float* c1w2  = (const float*)d_in[5];
  const float* c1b2  = (const float*)d_in[6];
  const float* c2w1  = (const float*)d_in[7];
  const float* c2b1  = (const float*)d_in[8];
  const float* c2w2  = (const float*)d_in[9];
  const float* c2b2  = (const float*)d_in[10];
  const float* hw0 = (const float*)d_in[11]; const float* hb0 = (const float*)d_in[12];
  const float* hw1 = (const float*)d_in[13]; const float* hb1 = (const float*)d_in[14];
  const float* hw2 = (const float*)d_in[15]; const float* hb2 = (const float*)d_in[16];
  const float* hw3 = (const float*)d_in[17]; const float* hb3 = (const float*)d_in[18];
  const float* hw4 = (const float*)d_in[19]; const float* hb4 = (const float*)d_in[20];

  // workspace layout
  char* ws = (char*)d_ws;
  const size_t nodeBytes = (size_t)N_NODES * HID * sizeof(float);  // 51.2 MB
  float*    agg    = (float*)ws;                          // also conv2 output
  float*    hbuf   = (float*)(ws + nodeBytes);            // conv1 output
  char*     small  = ws + 2 * nodeBytes;
  float*    psum   = (float*)small;                       small += NUM_GRAPHS * HID * 4;
  unsigned* pmax   = (unsigned*)small;                    small += NUM_GRAPHS * HID * 4;
  float*    pooled = (float*)small;                       small += NUM_GRAPHS * 2 * HID * 4;
  float*    g0     = (float*)small;                       small += NUM_GRAPHS * 128 * 4;
  float*    g1     = (float*)small;                       small += NUM_GRAPHS * 64 * 4;
  float*    g2     = (float*)small;                       small += NUM_GRAPHS * 32 * 4;
  float*    g3     = (float*)small;

  const int edgeBlocks = (int)(((size_t)N_EDGES * 32) / 256);   // 156250 (exact)
  const int ginBlocks  = N_NODES / 64;                          // 3125 (exact)
  const int poolBlocks = (int)(((size_t)N_NODES * HID) / 256);  // 50000 (exact)

  // ---- layer 1 ----
  hipMemsetAsync(agg, 0, nodeBytes, stream);
  scatter_add_kernel<<<edgeBlocks, 256, 0, stream>>>(x, src, dst, agg);
  gin_mlp_kernel<<<ginBlocks, 128, 0, stream>>>(x, agg, c1w1, c1b1, c1w2, c1b2,
                                                hbuf, /*relu_out=*/1);
  // ---- layer 2 ----
  hipMemsetAsync(agg, 0, nodeBytes, stream);
  scatter_add_kernel<<<edgeBlocks, 256, 0, stream>>>(hbuf, src, dst, agg);
  gin_mlp_kernel<<<ginBlocks, 128, 0, stream>>>(hbuf, agg, c2w1, c2b1, c2w2, c2b2,
                                                agg /*in-place h2*/, /*relu_out=*/0);
  // ---- pooling ----
  pool_init_kernel<<<(NUM_GRAPHS * HID + 255) / 256, 256, 0, stream>>>(psum, pmax);
  pool_kernel<<<poolBlocks, 256, 0, stream>>>(agg, batch, psum, pmax);
  pool_combine_kernel<<<(NUM_GRAPHS * 2 * HID + 255) / 256, 256, 0, stream>>>(psum, pmax, pooled);
  // ---- head MLP ----
  dense_kernel<<<NUM_GRAPHS, 128, 0, stream>>>(pooled, hw0, hb0, g0, 128, 128, 1);
  dense_kernel<<<NUM_GRAPHS,  64, 0, stream>>>(g0,     hw1, hb1, g1, 128,  64, 1);
  dense_kernel<<<NUM_GRAPHS,  32, 0, stream>>>(g1,     hw2, hb2, g2,  64,  32, 1);
  dense_kernel<<<NUM_GRAPHS,  32, 0, stream>>>(g2,     hw3, hb3, g3,  32,  16, 1);
  dense_kernel<<<NUM_GRAPHS,  32, 0, stream>>>(g3,     hw4, hb4, (float*)d_out, 16, 1, 0);
}